// SA_upsample_12481174962982
// MI455X (gfx1250) — compile-verified
//
#include <hip/hip_runtime.h>
#include <hip/hip_bf16.h>

// ---------------------------------------------------------------------------
// SA-upsample for MI455X (gfx1250, wave32, WMMA f32 16x16x4)
//
// Shapes (hardcoded from reference): B=4, C=64, H=W=128, E=4, scale=scale2=2
// Hh=Wh=256, Npix=65536.
//
// Kernel 1: per-pixel MLP (4->64->64->{2 offset, 4 routing}) using
//           V_WMMA_F32_16X16X4_F32, 16-pixel tiles per wave. Writes
//           routing[Npix][4], gridx[Npix], gridy[Npix] to d_ws.
// Kernel 2: bilinear sample -> fea0 written into d_out (B,64,256,256).
// Kernel 3: dynamic conv: mid = Sum_e (r_e * fea) @ Wc[e]^T   (K=64, N=8)
//                         out = Sum_e (r_e * mid) @ We[e]^T   (K=8,  N=64)
//           routing folded into A-fragments so per-pixel weights become
//           shared GEMMs; adds fea0 in place in d_out.
// ---------------------------------------------------------------------------

typedef __attribute__((ext_vector_type(2))) float v2f;
typedef __attribute__((ext_vector_type(8))) float v8f;

#define HH 256
#define WW 256
#define NPIX 65536     // HH*WW
#define HWIN 16384     // 128*128
#define INV127 (2.0f / 127.0f)

static __device__ inline v8f wmma4(v2f a, v2f b, v8f c) {
  // D = A(16x4) x B(4x16) + C(16x16), fp32
  return __builtin_amdgcn_wmma_f32_16x16x4_f32(
      /*neg_a=*/false, a, /*neg_b=*/false, b,
      /*c_mod=*/(short)0, c, /*reuse_a=*/false, /*reuse_b=*/false);
}

static __device__ inline v8f v8zero() {
  v8f z;
  for (int i = 0; i < 8; ++i) z[i] = 0.0f;
  return z;
}

// ---------------------------------------------------------------------------
// Kernel 1: MLP + routing + grid.  512 blocks x 256 threads (8 waves/block,
// each wave owns one 16-pixel tile; 4096 tiles total).
// ---------------------------------------------------------------------------
__global__ __launch_bounds__(256) void k_mlp(
    const float* __restrict__ w1, const float* __restrict__ b1,
    const float* __restrict__ w2, const float* __restrict__ b2,
    const float* __restrict__ wr, const float* __restrict__ br,
    const float* __restrict__ wo, const float* __restrict__ bo,
    float* __restrict__ rout, float* __restrict__ gridx,
    float* __restrict__ gridy) {
  __shared__ float sw1[64 * 4];
  __shared__ float sb1[64];
  __shared__ float sw2[64 * 64];
  __shared__ float sb2[64];
  __shared__ float swh[6 * 64];  // rows 0-1: wo, rows 2-5: wr
  __shared__ float sbh[8];
  __shared__ float semb[8][16][68];  // per-wave 16x64 tile, padded stride 68

  const int t = threadIdx.x;
  for (int i = t; i < 64 * 4; i += 256) sw1[i] = w1[i];
  for (int i = t; i < 64; i += 256) {
    sb1[i] = b1[i];
    sb2[i] = b2[i];
  }
  for (int i = t; i < 64 * 64; i += 256) sw2[i] = w2[i];
  for (int i = t; i < 6 * 64; i += 256)
    swh[i] = (i < 2 * 64) ? wo[i] : wr[i - 2 * 64];
  if (t < 6) sbh[t] = (t < 2) ? bo[t] : br[t - 2];
  __syncthreads();

  const int w = t >> 5;
  const int lane = t & 31;
  const int half = lane >> 4;  // 0: K=0,1 / rows M=i ; 1: K=2,3 / rows M=i+8
  const int l16 = lane & 15;
  const int klo = half * 2;
  const int tile = blockIdx.x * 8 + w;
  const int pbase = tile * 16;

  // ---- layer 1: emb1 = relu(inp(16x4) @ w1^T + b1) ----
  // A fragment: row = l16, K in {klo, klo+1}. inp channels:
  //   0: 1/scale2(=0.5), 1: 1/scale(=0.5), 2: coor_h(py), 3: coor_w(px)
  v2f a1;
  if (half == 0) {
    a1[0] = 0.5f;
    a1[1] = 0.5f;
  } else {
    const int p = pbase + l16;
    const int px = p & 255, py = p >> 8;
    const float th = (py + 0.5f) * 0.5f;
    const float tw = (px + 0.5f) * 0.5f;
    a1[0] = th - floorf(th + 0.001f) - 0.5f;  // coor_h
    a1[1] = tw - floorf(tw + 0.001f) - 0.5f;  // coor_w
  }
  for (int nt = 0; nt < 4; ++nt) {
    const int ng = nt * 16 + l16;
    v2f bf;
    bf[0] = sw1[ng * 4 + klo];
    bf[1] = sw1[ng * 4 + klo + 1];
    v8f c;
    const float bias = sb1[ng];
    for (int i = 0; i < 8; ++i) c[i] = bias;
    v8f d = wmma4(a1, bf, c);
    for (int i = 0; i < 8; ++i) {
      const float v = d[i] > 0.0f ? d[i] : 0.0f;
      semb[w][i + half * 8][ng] = v;  // (M = i + 8*half, col = ng)
    }
  }

  // ---- layer 2: emb2 = relu(emb1(16x64) @ w2^T + b2) ----
  v8f acc2[4];
  for (int nt = 0; nt < 4; ++nt) {
    const int ng = nt * 16 + l16;
    const float bias = sb2[ng];
    v8f c;
    for (int i = 0; i < 8; ++i) c[i] = bias;
    for (int kk = 0; kk < 16; ++kk) {
      const int kb = kk * 4 + klo;
      v2f av;
      av[0] = semb[w][l16][kb];
      av[1] = semb[w][l16][kb + 1];
      v2f bv;
      bv[0] = sw2[ng * 64 + kb];
      bv[1] = sw2[ng * 64 + kb + 1];
      c = wmma4(av, bv, c);
    }
    for (int i = 0; i < 8; ++i) c[i] = c[i] > 0.0f ? c[i] : 0.0f;
    acc2[nt] = c;
  }
  // overwrite semb with emb2 (all reads of emb1 are done; LDS in-order/wave)
  for (int nt = 0; nt < 4; ++nt) {
    const int ng = nt * 16 + l16;
    for (int i = 0; i < 8; ++i) semb[w][i + half * 8][ng] = acc2[nt][i];
  }

  // ---- heads: 6 outputs = emb2(16x64) @ whead^T + bhead (N padded to 16) ----
  v8f ch;
  {
    const float hb = (l16 < 6) ? sbh[l16] : 0.0f;
    for (int i = 0; i < 8; ++i) ch[i] = hb;
  }
  for (int kk = 0; kk < 16; ++kk) {
    const int kb = kk * 4 + klo;
    v2f av;
    av[0] = semb[w][l16][kb];
    av[1] = semb[w][l16][kb + 1];
    v2f bv;
    bv[0] = (l16 < 6) ? swh[l16 * 64 + kb] : 0.0f;
    bv[1] = (l16 < 6) ? swh[l16 * 64 + kb + 1] : 0.0f;
    ch = wmma4(av, bv, ch);
  }
  // extract: col 0 -> offset_x, col 1 -> offset_y, cols 2-5 -> routing
  for (int i = 0; i < 8; ++i) {
    const int m = i + half * 8;
    const int p = pbase + m;
    const int px = p & 255, py = p >> 8;
    const float v = ch[i];
    if (l16 == 0) {
      const float gxb = ((px + 0.5f) * 0.5f - 0.5f) * INV127 - 1.0f;
      gridx[p] = gxb + v * INV127;
    } else if (l16 == 1) {
      const float gyb = ((py + 0.5f) * 0.5f - 0.5f) * INV127 - 1.0f;
      gridy[p] = gyb + v * INV127;
    } else if (l16 < 6) {
      rout[p * 4 + (l16 - 2)] = 1.0f / (1.0f + __expf(-v));
    }
  }
}

// ---------------------------------------------------------------------------
// Kernel 2: bilinear sampling. One thread per (b, pixel); loops 64 channels.
// 1024 blocks x 256 threads. Writes fea0 into d_out.
// ---------------------------------------------------------------------------
__global__ __launch_bounds__(256) void k_sample(const float* __restrict__ x,
                                                const float* __restrict__ gridx,
                                                const float* __restrict__ gridy,
                                                float* __restrict__ out) {
  const int gid = blockIdx.x * 256 + threadIdx.x;
  const int b = gid >> 16;     // / NPIX
  const int p = gid & 65535;   // % NPIX
  const float gx = gridx[p];
  const float gy = gridy[p];
  const float ix = (gx + 1.0f) * 0.5f * 127.0f;
  const float iy = (gy + 1.0f) * 0.5f * 127.0f;
  const float ix0 = floorf(ix), iy0 = floorf(iy);
  const float wx = ix - ix0, wy = iy - iy0;

  const float cxf[4] = {ix0, ix0 + 1.0f, ix0, ix0 + 1.0f};
  const float cyf[4] = {iy0, iy0, iy0 + 1.0f, iy0 + 1.0f};
  const float cw[4] = {(1.0f - wx) * (1.0f - wy), wx * (1.0f - wy),
                       (1.0f - wx) * wy, wx * wy};
  int off[4];
  float wgt[4];
#pragma unroll
  for (int k = 0; k < 4; ++k) {
    const bool valid = (cxf[k] >= 0.0f) && (cxf[k] <= 127.0f) &&
                       (cyf[k] >= 0.0f) && (cyf[k] <= 127.0f);
    const int xi = (int)fminf(fmaxf(cxf[k], 0.0f), 127.0f);
    const int yi = (int)fminf(fmaxf(cyf[k], 0.0f), 127.0f);
    off[k] = yi * 128 + xi;
    wgt[k] = valid ? cw[k] : 0.0f;
  }
  const float* xb = x + (size_t)b * 64 * HWIN;
  float* ob = out + (size_t)b * 64 * NPIX + p;
#pragma unroll 4
  for (int c = 0; c < 64; ++c) {
    const float* xc = xb + c * HWIN;
    const float v = wgt[0] * xc[off[0]] + wgt[1] * xc[off[1]] +
                    wgt[2] * xc[off[2]] + wgt[3] * xc[off[3]];
    ob[(size_t)c * NPIX] = v;
  }
}

// ---------------------------------------------------------------------------
// Kernel 3: routing-folded dynamic conv. 2048 blocks x 256 threads.
// Each wave: one (batch, 16-pixel tile).
// ---------------------------------------------------------------------------
__global__ __launch_bounds__(256) void k_apply(const float* __restrict__ wc,
                                               const float* __restrict__ we,
                                               const float* __restrict__ rout,
                                               float* __restrict__ out) {
  __shared__ float sWc[4 * 512];      // (E,8,64)
  __shared__ float sWe[4 * 512];      // (E,64,8)
  __shared__ float sfea[8][16][68];   // per-wave fea0 tile (padded)
  __shared__ float smid[8][16][12];   // per-wave mid tile (8 cols, padded)
  __shared__ float srout[8][16][4];   // per-wave routing tile

  const int t = threadIdx.x;
  for (int i = t; i < 2048; i += 256) {
    sWc[i] = wc[i];
    sWe[i] = we[i];
  }
  __syncthreads();

  const int w = t >> 5;
  const int lane = t & 31;
  const int half = lane >> 4;
  const int l16 = lane & 15;
  const int klo = half * 2;
  const int g = blockIdx.x * 8 + w;
  const int b = g >> 12;      // / 4096
  const int tile = g & 4095;  // % 4096
  const size_t base = (size_t)b * 64 * NPIX + (size_t)tile * 16;

  // stage fea0 tile + routing tile (coalesced over 16 contiguous pixels)
  for (int idx = lane; idx < 1024; idx += 32) {
    const int c = idx >> 4, m = idx & 15;
    sfea[w][m][c] = out[base + (size_t)c * NPIX + m];
  }
  for (int idx = lane; idx < 64; idx += 32) {
    const int m = idx >> 2, e = idx & 3;
    srout[w][m][e] = rout[(tile * 16 + m) * 4 + e];
  }
  __syncthreads();

  // ---- GEMM 1: mid(16x8) = Sum_e (r_e * fea)(16x64) @ Wc[e]^T(64x8pad16) ----
  v8f accM = v8zero();
  for (int e = 0; e < 4; ++e) {
    const float re = srout[w][l16][e];
    for (int kk = 0; kk < 16; ++kk) {
      const int kb = kk * 4 + klo;
      v2f av;
      av[0] = re * sfea[w][l16][kb];
      av[1] = re * sfea[w][l16][kb + 1];
      v2f bv;
      bv[0] = (l16 < 8) ? sWc[e * 512 + l16 * 64 + kb] : 0.0f;
      bv[1] = (l16 < 8) ? sWc[e * 512 + l16 * 64 + kb + 1] : 0.0f;
      accM = wmma4(av, bv, accM);
    }
  }
  if (l16 < 8) {
    for (int i = 0; i < 8; ++i) smid[w][i + half * 8][l16] = accM[i];
  }

  // ---- GEMM 2: out(16x64) = Sum_e (r_e * mid)(16x8) @ We[e]^T(8x64) ----
  v8f accO[4];
  for (int nt = 0; nt < 4; ++nt) accO[nt] = v8zero();
  for (int e = 0; e < 4; ++e) {
    const float re = srout[w][l16][e];
    for (int kk = 0; kk < 2; ++kk) {
      const int kb = kk * 4 + klo;
      v2f av;
      av[0] = re * smid[w][l16][kb];
      av[1] = re * smid[w][l16][kb + 1];
      for (int nt = 0; nt < 4; ++nt) {
        const int ng = nt * 16 + l16;
        v2f bv;
        bv[0] = sWe[e * 512 + ng * 8 + kb];
        bv[1] = sWe[e * 512 + ng * 8 + kb + 1];
        accO[nt] = wmma4(av, bv, accO[nt]);
      }
    }
  }

  // ---- add fea0 (residual) into LDS, then coalesced store ----
  for (int nt = 0; nt < 4; ++nt) {
    const int ng = nt * 16 + l16;
    for (int i = 0; i < 8; ++i) {
      const int m = i + half * 8;
      sfea[w][m][ng] = accO[nt][i] + sfea[w][m][ng];
    }
  }
  for (int idx = lane; idx < 1024; idx += 32) {
    const int c = idx >> 4, m = idx & 15;
    out[base + (size_t)c * NPIX + m] = sfea[w][m][c];
  }
}

// ---------------------------------------------------------------------------
// Host launcher
// ---------------------------------------------------------------------------
extern "C" void kernel_launch(void* const* d_in, const int* in_sizes, int n_in,
                              void* d_out, int out_size, void* d_ws,
                              size_t ws_size, hipStream_t stream) {
  (void)in_sizes;
  (void)n_in;
  (void)out_size;
  (void)ws_size;

  const float* x = (const float*)d_in[0];    // (4,64,128,128)
  const float* wcm = (const float*)d_in[1];  // (4,8,64)
  const float* wem = (const float*)d_in[2];  // (4,64,8)
  const float* w1 = (const float*)d_in[3];   // (64,4)
  const float* b1 = (const float*)d_in[4];   // (64)
  const float* w2 = (const float*)d_in[5];   // (64,64)
  const float* b2 = (const float*)d_in[6];   // (64)
  const float* wr = (const float*)d_in[7];   // (4,64)
  const float* br = (const float*)d_in[8];   // (4)
  const float* wo = (const float*)d_in[9];   // (2,64)
  const float* bo = (const float*)d_in[10];  // (2)
  // d_in[11], d_in[12]: scale=2, scale2=2 (hardcoded)

  float* ws = (float*)d_ws;
  float* rout = ws;                  // [NPIX][4]
  float* gridx = ws + 4 * NPIX;      // [NPIX]
  float* gridy = gridx + NPIX;       // [NPIX]
  float* out = (float*)d_out;        // (4,64,256,256)

  // 4096 pixel tiles, 8 waves/block -> 512 blocks
  k_mlp<<<512, 256, 0, stream>>>(w1, b1, w2, b2, wr, br, wo, bo, rout, gridx,
                                 gridy);
  // 4*65536 threads -> 1024 blocks
  k_sample<<<1024, 256, 0, stream>>>(x, gridx, gridy, out);
  // 4*4096 wave-tiles, 8 waves/block -> 2048 blocks
  k_apply<<<2048, 256, 0, stream>>>(wcm, wem, rout, out);
}